// MLASelfAttn_72584947302731
// MI455X (gfx1250) — compile-verified
//
#include <hip/hip_runtime.h>

typedef __attribute__((ext_vector_type(2))) float v2f;
typedef __attribute__((ext_vector_type(8))) float v8f;
typedef int v4i_vs __attribute__((vector_size(16)));   // matches async-builtin param type

#define TOKENS   8192          // B*S = 2*4096
#define OUT_W    6720          // 32*192 + 64 + 512
#define ROPE_D   64

#define MBLK 128
#define NBLK 64
#define KBLK 16
#define ASTR 20                // padded LDS row stride (16 data + 4 pad): conflict-free ds_load_b64 frags
#define NBUF 3                 // triple buffer -> 2 async stages in flight

// ---- gfx1250 async memory->LDS path (guarded; falls back to sync copy) ----
#if defined(__has_builtin)
#if __has_builtin(__builtin_amdgcn_global_load_async_to_lds_b128)
#define HAVE_ASYNC_LDS 1
#endif
#if __has_builtin(__builtin_amdgcn_s_wait_asynccnt)
#define WAIT_ASYNC_LE(n) __builtin_amdgcn_s_wait_asynccnt(n)
#else
#define WAIT_ASYNC_LE(n) asm volatile("s_wait_asynccnt %0" :: "i"(n) : "memory")
#endif
#else
#define WAIT_ASYNC_LE(n) asm volatile("s_wait_asynccnt %0" :: "i"(n) : "memory")
#endif

// C[m][n] = sum_k A[m][k] * W[n][k]    A: MxK row-major, W: NxK row-major (B = W^T)
// mode 0: C[m*ldc + n]
// mode 1: kv remap: n<512 -> compressed_kv at col n+6208 ; n in [512,576) -> k_pe at col n+5632
__global__ __launch_bounds__(256)
void mla_gemm_f32_wmma(const float* __restrict__ A, const float* __restrict__ W,
                       float* __restrict__ C, int M, int N, int K, int ldc, int mode)
{
    // triple-buffered tiles; both stored row-major (matrix row = LDS row, stride 20)
    __shared__ __align__(16) float As[NBUF][MBLK * ASTR];   // 3 x 128 x 20
    __shared__ __align__(16) float Bs[NBUF][NBLK * ASTR];   // 3 x  64 x 20

    const int tid  = threadIdx.x;
    const int lane = tid & 31;
    const int wave = tid >> 5;          // 0..7
    const int wm   = wave & 3;          // 4 waves along M -> 32-row slabs
    const int wn   = wave >> 2;         // 2 waves along N -> 32-col slabs

    const int m0 = blockIdx.y * MBLK;
    const int n0 = blockIdx.x * NBLK;

    const int lrow = lane & 15;         // M (A frag) / N (B frag) index within 16
    const int lhi  = lane >> 4;         // 0 -> K={0,1}, 1 -> K={2,3}

    v8f acc[2][2];
#pragma unroll
    for (int i = 0; i < 2; ++i)
#pragma unroll
        for (int j = 0; j < 2; ++j)
#pragma unroll
            for (int v = 0; v < 8; ++v)
                acc[i][j][v] = 0.0f;

    // ---- per-thread staging addresses (computed once, bumped by KBLK per stage) ----
    // A tile 128x16 = 512 float4 chunks: copy0 rows 0..63, copy1 rows 64..127
    // B tile  64x16 = 256 float4 chunks: same (row,col) mapping as A copy0
    const int srow   = tid >> 2;              // 0..63
    const int scol   = (tid & 3) << 2;        // 0,4,8,12
    const int ldsOfs = srow * ASTR + scol;    // shared by A-copy0 and B-copy
    const float* gA = A + (size_t)(m0 + srow) * K + scol;
    const float* gW = W + (size_t)(n0 + srow) * K + scol;
    const size_t gA2 = (size_t)64 * K;        // A-copy1 source offset (rows +64)

    // issue one K-tile stage (3 async b128 per thread), then advance source pointers
    auto stage = [&](float* aDst, float* bDst) {
#if HAVE_ASYNC_LDS
        __builtin_amdgcn_global_load_async_to_lds_b128(
            (v4i_vs*)gA,         (v4i_vs*)(aDst + ldsOfs), 0, 0);
        __builtin_amdgcn_global_load_async_to_lds_b128(
            (v4i_vs*)(gA + gA2), (v4i_vs*)(aDst + ldsOfs + 64 * ASTR), 0, 0);
        __builtin_amdgcn_global_load_async_to_lds_b128(
            (v4i_vs*)gW,         (v4i_vs*)(bDst + ldsOfs), 0, 0);
#else
        *(float4*)(aDst + ldsOfs)             = *(const float4*)gA;
        *(float4*)(aDst + ldsOfs + 64 * ASTR) = *(const float4*)(gA + gA2);
        *(float4*)(bDst + ldsOfs)             = *(const float4*)gW;
#endif
        gA += KBLK;
        gW += KBLK;
    };

    // 16 x v_wmma_f32_16x16x4_f32 on the wave's 32x32 tile; A and B frags are
    // identical conflict-free ds_load_b64 patterns (row*20 + k)
    auto compute = [&](const float* aBase, const float* bBase) {
#pragma unroll
        for (int ks = 0; ks < KBLK; ks += 4) {
            v2f a[2], b[2];
#pragma unroll
            for (int i = 0; i < 2; ++i) {
                int arow = wm * 32 + i * 16 + lrow;
                a[i] = *(const v2f*)(aBase + arow * ASTR + ks + 2 * lhi);
            }
#pragma unroll
            for (int j = 0; j < 2; ++j) {
                int brow = wn * 32 + j * 16 + lrow;
                b[j] = *(const v2f*)(bBase + brow * ASTR + ks + 2 * lhi);
            }
#pragma unroll
            for (int i = 0; i < 2; ++i)
#pragma unroll
                for (int j = 0; j < 2; ++j)
                    acc[i][j] = __builtin_amdgcn_wmma_f32_16x16x4_f32(
                        false, a[i], false, b[j], (short)0, acc[i][j], false, false);
        }
    };

    // ---- 2-deep async pipeline, branch-free steady state, rotated buffer pointers ----
    float *a0 = &As[0][0], *a1 = &As[1][0], *a2 = &As[2][0];
    float *b0 = &Bs[0][0], *b1 = &Bs[1][0], *b2 = &Bs[2][0];

    const int T = K / KBLK;               // 256 / 96 / 256 for the three GEMMs
    stage(a0, b0);
    if (T > 1) {
        stage(a1, b1);
        for (int t = 0; t < T - 2; ++t) {
            WAIT_ASYNC_LE(3);             // oldest stage retired (in-order completion)
            __syncthreads();              // all waves: stage t done, tile t-1 reads drained
            stage(a2, b2);                // overwrite buffer read last iteration (tile t-1)
            compute(a0, b0);
            float* ta = a0; a0 = a1; a1 = a2; a2 = ta;
            float* tb = b0; b0 = b1; b1 = b2; b2 = tb;
        }
        WAIT_ASYNC_LE(3);                 // tile T-2 ready (stage T-1 still in flight)
        __syncthreads();
        compute(a0, b0);
        { float* ta = a0; a0 = a1; a1 = ta; float* tb = b0; b0 = b1; b1 = tb; }
    }
    WAIT_ASYNC_LE(0);                     // last tile ready
    __syncthreads();
    compute(a0, b0);

    // epilogue: lane holds col (lane&15); vgpr v holds row v + 8*(lane>=16)
#pragma unroll
    for (int i = 0; i < 2; ++i)
#pragma unroll
        for (int j = 0; j < 2; ++j) {
            int row = m0 + wm * 32 + i * 16 + 8 * lhi;
            int col = n0 + wn * 32 + j * 16 + lrow;
            int oc  = mode ? (col < 512 ? col + 6208 : col + 5632) : col;
            float* cp = C + (size_t)row * ldc + oc;
#pragma unroll
            for (int v = 0; v < 8; ++v)
                cp[(size_t)v * ldc] = acc[i][j][v];
        }
}

// In-place RoPE on the pe slices of d_out. Each thread owns a (j, j+32) pair -> race-free,
// and the GEMMs fully rewrite these columns every call -> deterministic under graph replay.
// "head" 0..31 = q heads (cols h*192+128 .. +191), "head" 32 = k_pe (cols 6144..6207).
__global__ __launch_bounds__(256)
void mla_rope_inplace(float* __restrict__ out,
                      const float* __restrict__ cs, const float* __restrict__ sn)
{
    const int tok = blockIdx.x;
    float* row = out + (size_t)tok * OUT_W;
    const float* c = cs + (size_t)tok * ROPE_D;
    const float* s = sn + (size_t)tok * ROPE_D;

    for (int w = threadIdx.x; w < 33 * 32; w += 256) {
        int h = w >> 5;
        int j = w & 31;
        int base = (h < 32) ? (h * 192 + 128) : 6144;
        float a = row[base + j];
        float b = row[base + j + 32];
        // out[j] = t[j]*c[j] - t[j+32]*s[j] ; out[j+32] = t[j+32]*c[j+32] + t[j]*s[j+32]
        row[base + j]      = a * c[j]      - b * s[j];
        row[base + j + 32] = b * c[j + 32] + a * s[j + 32];
    }
}

extern "C" void kernel_launch(void* const* d_in, const int* in_sizes, int n_in,
                              void* d_out, int out_size, void* d_ws, size_t ws_size,
                              hipStream_t stream) {
    const float* x    = (const float*)d_in[0];   // [2,4096,4096]
    const float* cosp = (const float*)d_in[1];   // [2,4096,64]
    const float* sinp = (const float*)d_in[2];   // [2,4096,64]
    const float* Wqa  = (const float*)d_in[3];   // [1536,4096]
    const float* Wqb  = (const float*)d_in[4];   // [6144,1536]
    const float* Wkva = (const float*)d_in[5];   // [576,4096]
    float* out   = (float*)d_out;                // [8192, 6720]
    float* qlora = (float*)d_ws;                 // [8192, 1536] f32 = 48 MB scratch

    dim3 blk(256);

    // GEMM1: q_lora = x @ Wqa^T            M=8192 N=1536 K=4096
    mla_gemm_f32_wmma<<<dim3(1536 / NBLK, TOKENS / MBLK), blk, 0, stream>>>(
        x, Wqa, qlora, TOKENS, 1536, 4096, 1536, 0);

    // GEMM2: q = q_lora @ Wqb^T -> out cols [0,6144)   M=8192 N=6144 K=1536
    mla_gemm_f32_wmma<<<dim3(6144 / NBLK, TOKENS / MBLK), blk, 0, stream>>>(
        qlora, Wqb, out, TOKENS, 6144, 1536, OUT_W, 0);

    // GEMM3: kv = x @ Wkva^T -> out cols [6144,6720) with kv/k_pe remap   M=8192 N=576 K=4096
    mla_gemm_f32_wmma<<<dim3(576 / NBLK, TOKENS / MBLK), blk, 0, stream>>>(
        x, Wkva, out, TOKENS, 576, 4096, OUT_W, 1);

    // RoPE in place on q_pe (32 heads) + k_pe
    mla_rope_inplace<<<dim3(TOKENS), blk, 0, stream>>>(out, cosp, sinp);
}